// GCNLayer_88295937671172
// MI455X (gfx1250) — compile-verified
//
#include <hip/hip_runtime.h>
#include <math.h>

typedef __attribute__((ext_vector_type(2))) float v2f;
typedef __attribute__((ext_vector_type(4))) float v4f;
typedef __attribute__((ext_vector_type(8))) float v8f;

#define DIM 128
#define LEAKY 0.01f
#define BN_EPS 1e-5f

__device__ __forceinline__ float warpReduceMax(float v) {
#pragma unroll
  for (int off = 16; off >= 1; off >>= 1) v = fmaxf(v, __shfl_xor(v, off, 32));
  return v;
}
__device__ __forceinline__ float warpReduceSum(float v) {
#pragma unroll
  for (int off = 16; off >= 1; off >>= 1) v += __shfl_xor(v, off, 32);
  return v;
}

// out[i][j] = act( in[i][:] ) . W[j][:]  (+ bias[j])
// FUSE_BN: act(v,k) = leaky( v*bnScale[k] + bnShift[k] )
// DO_STATS: accumulate per-feature sum / sumsq of the OUTPUT into gStats[0:128]/[128:256]
template <bool FUSE_BN, bool DO_STATS>
__global__ void __launch_bounds__(256)
gemm_wmma(const float* __restrict__ in, const float* __restrict__ W,
          const float* __restrict__ bias,
          const float* __restrict__ bnScale, const float* __restrict__ bnShift,
          float* __restrict__ out, float* __restrict__ gStats, int nrows) {
  __shared__ float sW[DIM * DIM];     // 64 KB weight tile (whole weight matrix)
  __shared__ float sAff[2 * DIM];     // BN scale | shift
  __shared__ float sStat[2 * DIM];    // block-local sum | sumsq

  const int tid = threadIdx.x;

  // cooperative weight load, float4 wide
  {
    const v4f* src = (const v4f*)W;
    v4f* dst = (v4f*)sW;
    for (int i = tid; i < DIM * DIM / 4; i += 256) dst[i] = src[i];
  }
  if (FUSE_BN && tid < DIM) {
    sAff[tid] = bnScale[tid];
    sAff[DIM + tid] = bnShift[tid];
  }
  if (DO_STATS) sStat[tid] = 0.f;  // 256 threads cover 256 entries
  __syncthreads();

  const int lane = tid & 31;
  const int wave = tid >> 5;
  const int l15 = lane & 15;
  const int hi = lane >> 4;                 // half-wave select (K split / M split)
  const int m0 = blockIdx.x * 128 + wave * 16;

  int rowA = m0 + l15;
  if (rowA >= nrows) rowA = nrows - 1;      // clamp loads; stores are guarded
  const float* __restrict__ pin = in + (size_t)rowA * DIM;

  const v8f vzero = {0.f, 0.f, 0.f, 0.f, 0.f, 0.f, 0.f, 0.f};
  v8f acc[8];
#pragma unroll
  for (int t = 0; t < 8; ++t) acc[t] = vzero;

  // K = 128 in steps of 4 (V_WMMA_F32_16X16X4_F32)
  for (int k0 = 0; k0 < DIM; k0 += 4) {
    const int kk = k0 + 2 * hi;             // lanes 0-15: K=k0,k0+1 ; lanes 16-31: K=k0+2,k0+3
    v2f a = *(const v2f*)(pin + kk);
    if (FUSE_BN) {
      float ax = a.x * sAff[kk] + sAff[DIM + kk];
      float ay = a.y * sAff[kk + 1] + sAff[DIM + kk + 1];
      a.x = ax >= 0.f ? ax : LEAKY * ax;
      a.y = ay >= 0.f ? ay : LEAKY * ay;
    }
#pragma unroll
    for (int t = 0; t < 8; ++t) {
      const int n = t * 16 + l15;           // B: K rows kk,kk+1 of column n (W row-major: W[n][k])
      v2f b = *(const v2f*)(sW + n * DIM + kk);
      acc[t] = __builtin_amdgcn_wmma_f32_16x16x4_f32(
          false, a, false, b, (short)0, acc[t], false, false);
    }
  }

  // C/D layout: VGPR r, lanes 0-15 -> M=r, lanes 16-31 -> M=r+8; N = l15 (+ tile offset)
#pragma unroll
  for (int t = 0; t < 8; ++t) {
    const int col = t * 16 + l15;
    const float bb = bias ? bias[col] : 0.f;
    float ps = 0.f, pq = 0.f;
#pragma unroll
    for (int r = 0; r < 8; ++r) {
      const int row = m0 + hi * 8 + r;
      const float v = acc[t][r] + bb;
      if (row < nrows) {
        out[(size_t)row * DIM + col] = v;
        ps += v;
        pq += v * v;
      }
    }
    if (DO_STATS) {
      atomicAdd(&sStat[col], ps);           // ds_add_f32
      atomicAdd(&sStat[DIM + col], pq);
    }
  }
  if (DO_STATS) {
    __syncthreads();
    unsafeAtomicAdd(&gStats[tid], sStat[tid]);  // one global f32 atomic per feature per block
  }
}

__global__ void bn_stats(const float* __restrict__ gStats,
                         const float* __restrict__ gamma, const float* __restrict__ beta,
                         float* __restrict__ bnScale, float* __restrict__ bnShift, int nrows) {
  const int j = threadIdx.x;  // 128 threads
  const float invN = 1.f / (float)nrows;
  const float mean = gStats[j] * invN;
  const float var = gStats[DIM + j] * invN - mean * mean;
  const float rinv = rsqrtf(var + BN_EPS);
  const float sc = gamma[j] * rinv;
  bnScale[j] = sc;
  bnShift[j] = beta[j] - mean * sc;
}

__global__ void deg_init(float* __restrict__ deg, int n) {
  const int i = blockIdx.x * 256 + threadIdx.x;
  if (i < n) deg[i] = 1.f;  // self loop
}

__global__ void deg_count(const int* __restrict__ dst, float* __restrict__ deg, int ne) {
  const int e = blockIdx.x * 256 + threadIdx.x;
  if (e < ne) unsafeAtomicAdd(&deg[dst[e]], 1.f);
}

__global__ void deg_rsqrt(float* __restrict__ deg, int n) {
  const int i = blockIdx.x * 256 + threadIdx.x;
  if (i < n) deg[i] = rsqrtf(deg[i]);  // deg >= 1 always -> dinv in place
}

// one wave per edge: 32 lanes x float4 = 128 features
__global__ void __launch_bounds__(256)
scatter_edges(const int* __restrict__ src, const int* __restrict__ dst,
              const float* __restrict__ h, const float* __restrict__ dinv,
              float* __restrict__ out, int ne) {
  const int lane = threadIdx.x & 31;
  const int e = blockIdx.x * 8 + (threadIdx.x >> 5);
  if (e >= ne) return;
  const int s = src[e];
  const int d = dst[e];
  const float norm = dinv[s] * dinv[d];
  const v4f v = *(const v4f*)(h + (size_t)s * DIM + lane * 4);
  float* po = out + (size_t)d * DIM + lane * 4;
  unsafeAtomicAdd(po + 0, norm * v.x);
  unsafeAtomicAdd(po + 1, norm * v.y);
  unsafeAtomicAdd(po + 2, norm * v.z);
  unsafeAtomicAdd(po + 3, norm * v.w);
}

// one wave per row: self-loop term + bias + log_softmax, in place
__global__ void __launch_bounds__(256)
finalize_rows(const float* __restrict__ h, const float* __restrict__ dinv,
              const float* __restrict__ bg, float* __restrict__ out, int n) {
  const int lane = threadIdx.x & 31;
  const int row = blockIdx.x * 8 + (threadIdx.x >> 5);
  if (row >= n) return;
  const float di = dinv[row];
  const float sc = di * di;
  float* po = out + (size_t)row * DIM + lane * 4;
  v4f v = *(v4f*)po;
  const v4f hv = *(const v4f*)(h + (size_t)row * DIM + lane * 4);
  const v4f bgv = *(const v4f*)(bg + lane * 4);
  v.x += sc * hv.x + bgv.x;
  v.y += sc * hv.y + bgv.y;
  v.z += sc * hv.z + bgv.z;
  v.w += sc * hv.w + bgv.w;
  float m = fmaxf(fmaxf(v.x, v.y), fmaxf(v.z, v.w));
  m = warpReduceMax(m);
  float s = expf(v.x - m) + expf(v.y - m) + expf(v.z - m) + expf(v.w - m);
  s = warpReduceSum(s);
  const float lse = m + logf(s);
  v.x -= lse;
  v.y -= lse;
  v.z -= lse;
  v.w -= lse;
  *(v4f*)po = v;
}

extern "C" void kernel_launch(void* const* d_in, const int* in_sizes, int n_in,
                              void* d_out, int out_size, void* d_ws, size_t ws_size,
                              hipStream_t stream) {
  const float* fea   = (const float*)d_in[0];
  const int*   edges = (const int*)d_in[1];
  const float* Wlin  = (const float*)d_in[2];
  const float* blin  = (const float*)d_in[3];
  const float* gamma = (const float*)d_in[4];
  const float* beta  = (const float*)d_in[5];
  const float* Wg    = (const float*)d_in[6];
  const float* bg    = (const float*)d_in[7];
  float* out = (float*)d_out;

  const int n  = in_sizes[0] / DIM;   // nodes
  const int ne = in_sizes[1] / 2;     // edges
  const int* esrc = edges;
  const int* edst = edges + ne;

  // workspace layout
  char* ws = (char*)d_ws;
  size_t off = 0;
  float* h = (float*)(ws + off);
  off += (size_t)n * DIM * sizeof(float);
  float* dinv = (float*)(ws + off);
  off += (((size_t)n * sizeof(float)) + 255) & ~(size_t)255;
  float* gStats = (float*)(ws + off);
  off += 256 * sizeof(float);
  float* bnScale = (float*)(ws + off);
  off += DIM * sizeof(float);
  float* bnShift = (float*)(ws + off);
  off += DIM * sizeof(float);

  const int mBlocks = (n + 127) / 128;

  // Phase 1: x = fea @ Wlin^T + blin  (x stored in d_out), fused BN stats
  hipMemsetAsync(gStats, 0, 256 * sizeof(float), stream);
  gemm_wmma<false, true><<<mBlocks, 256, 0, stream>>>(
      fea, Wlin, blin, nullptr, nullptr, out, gStats, n);

  // Phase 2: fold BN into affine per feature
  bn_stats<<<1, 128, 0, stream>>>(gStats, gamma, beta, bnScale, bnShift, n);

  // Phase 3: h = leaky(BN(x)) @ Wg^T  (reads x from d_out, writes h to ws)
  gemm_wmma<true, false><<<mBlocks, 256, 0, stream>>>(
      out, Wg, nullptr, bnScale, bnShift, h, nullptr, n);

  // Phase 4: degrees -> dinv (in place)
  deg_init<<<(n + 255) / 256, 256, 0, stream>>>(dinv, n);
  deg_count<<<(ne + 255) / 256, 256, 0, stream>>>(edst, dinv, ne);
  deg_rsqrt<<<(n + 255) / 256, 256, 0, stream>>>(dinv, n);

  // Phase 5: edge scatter into zeroed d_out
  hipMemsetAsync(out, 0, (size_t)n * DIM * sizeof(float), stream);
  scatter_edges<<<(ne + 7) / 8, 256, 0, stream>>>(esrc, edst, h, dinv, out, ne);

  // Phase 6: self loop + bias + log_softmax (in place)
  finalize_rows<<<(n + 7) / 8, 256, 0, stream>>>(h, dinv, bg, out, n);
}